// GSNET_81535659147320
// MI455X (gfx1250) — compile-verified
//
#include <hip/hip_runtime.h>
#include <float.h>
#include <math.h>

#define KNN 20
#define SLOPE 0.2f
#define BN_INVS 0.99999500003749937f   // 1/sqrt(1+1e-5)

typedef float v2f __attribute__((ext_vector_type(2)));
typedef float v8f __attribute__((ext_vector_type(8)));

// ---------- helpers ----------
__device__ __forceinline__ int f2ord(float f) {
    int i = __float_as_int(f);
    return (i >= 0) ? i : (i ^ 0x7fffffff);
}
__device__ __forceinline__ float ord2f(int i) {
    return __int_as_float((i >= 0) ? i : (i ^ 0x7fffffff));
}

// closed-form ascending eigenvalues of symmetric 3x3
__device__ void sym3_eigvals(float a11, float a12, float a13,
                             float a22, float a23, float a33, float ev[3]) {
    float p1 = a12*a12 + a13*a13 + a23*a23;
    float q  = (a11 + a22 + a33) * (1.0f/3.0f);
    float d1 = a11 - q, d2 = a22 - q, d3 = a33 - q;
    float p2 = d1*d1 + d2*d2 + d3*d3 + 2.0f*p1;
    if (p2 <= 1e-30f) { ev[0] = ev[1] = ev[2] = q; return; }
    float p = sqrtf(p2 * (1.0f/6.0f));
    float inv = 1.0f / p;
    float b11 = d1*inv, b12 = a12*inv, b13 = a13*inv;
    float b22 = d2*inv, b23 = a23*inv, b33 = d3*inv;
    float detB = b11*(b22*b33 - b23*b23) - b12*(b12*b33 - b23*b13)
               + b13*(b12*b23 - b22*b13);
    float r = 0.5f * detB;
    r = fminf(1.0f, fmaxf(-1.0f, r));
    float phi = acosf(r) * (1.0f/3.0f);
    float e3 = q + 2.0f*p*cosf(phi);
    float e1 = q + 2.0f*p*cosf(phi + 2.0943951023931953f);
    float e2 = 3.0f*q - e1 - e3;
    ev[0] = e1; ev[1] = e2; ev[2] = e3;
}

// ---------- KNN on 3-channel features: one block per (b,n) row ----------
__global__ __launch_bounds__(256)
void knn3_kernel(const float* __restrict__ f, int* __restrict__ idx, int N) {
    const int b = blockIdx.y, n = blockIdx.x, tid = threadIdx.x;
    __shared__ float val[1024];
    __shared__ float rv[256];
    __shared__ int   ri[256];
    const float x0 = f[(b*3+0)*N + n];
    const float x1 = f[(b*3+1)*N + n];
    const float x2 = f[(b*3+2)*N + n];
    for (int m = tid; m < N; m += 256) {
        float d0 = f[(b*3+0)*N + m] - x0;
        float d1 = f[(b*3+1)*N + m] - x1;
        float d2 = f[(b*3+2)*N + m] - x2;
        val[m] = -(d0*d0 + d1*d1 + d2*d2);
    }
    __syncthreads();
    for (int kk = 0; kk < KNN; ++kk) {
        float best = -FLT_MAX; int bi = 0x7fffffff;
        for (int m = tid; m < N; m += 256) {
            float v = val[m];
            if (v > best) { best = v; bi = m; }
        }
        rv[tid] = best; ri[tid] = bi;
        __syncthreads();
        for (int s = 128; s > 0; s >>= 1) {
            if (tid < s) {
                float vo = rv[tid+s]; int io = ri[tid+s];
                if (vo > rv[tid] || (vo == rv[tid] && io < ri[tid])) {
                    rv[tid] = vo; ri[tid] = io;
                }
            }
            __syncthreads();
        }
        if (tid == 0) { idx[(b*N + n)*KNN + kk] = ri[0]; val[ri[0]] = -FLT_MAX; }
        __syncthreads();
    }
}

// ---------- covariance eigenvalues per point ----------
__global__ void eig_kernel(const float* __restrict__ pts, const int* __restrict__ idx,
                           float* __restrict__ eig, int N, int B) {
    int t = blockIdx.x * blockDim.x + threadIdx.x;
    if (t >= B * N) return;
    int b = t / N, n = t % N;
    float cx = pts[(b*3+0)*N + n], cy = pts[(b*3+1)*N + n], cz = pts[(b*3+2)*N + n];
    float a11=0, a12=0, a13=0, a22=0, a23=0, a33=0;
    for (int k = 0; k < KNN; ++k) {
        int e = idx[(b*N + n)*KNN + k];
        float dx = pts[(b*3+0)*N + e] - cx;
        float dy = pts[(b*3+1)*N + e] - cy;
        float dz = pts[(b*3+2)*N + e] - cz;
        a11 += dx*dx; a12 += dx*dy; a13 += dx*dz;
        a22 += dy*dy; a23 += dy*dz; a33 += dz*dz;
    }
    float ev[3]; sym3_eigvals(a11, a12, a13, a22, a23, a33, ev);
    eig[(b*3+0)*N + n] = ev[0];
    eig[(b*3+1)*N + n] = ev[1];
    eig[(b*3+2)*N + n] = ev[2];
}

// ---------- conv1: 13->64 edge conv + BN + lrelu + max over k ----------
__global__ __launch_bounds__(64)
void conv1_kernel(const float* __restrict__ pts, const float* __restrict__ eig,
                  const int* __restrict__ idx_eu, const int* __restrict__ idx_ei,
                  const float* __restrict__ W, const float* __restrict__ bng,
                  const float* __restrict__ bnb, float* __restrict__ out, int N) {
    const int b = blockIdx.y, n = blockIdx.x, tid = threadIdx.x;
    __shared__ float G[KNN][14];
    if (tid < KNN) {
        int eu = idx_eu[(b*N + n)*KNN + tid];
        int ei = idx_ei[(b*N + n)*KNN + tid];
        float cx = pts[(b*3+0)*N + n],  cy = pts[(b*3+1)*N + n],  cz = pts[(b*3+2)*N + n];
        float nx = pts[(b*3+0)*N + eu], ny = pts[(b*3+1)*N + eu], nz = pts[(b*3+2)*N + eu];
        G[tid][0] = nx - cx; G[tid][1] = ny - cy; G[tid][2] = nz - cz;
        G[tid][3] = nx;      G[tid][4] = ny;      G[tid][5] = nz;
        float e0 = eig[(b*3+0)*N + n],  e1 = eig[(b*3+1)*N + n],  e2 = eig[(b*3+2)*N + n];
        float f0 = eig[(b*3+0)*N + ei], f1 = eig[(b*3+1)*N + ei], f2 = eig[(b*3+2)*N + ei];
        G[tid][6] = f0 - e0; G[tid][7] = f1 - e1; G[tid][8] = f2 - e2;
        G[tid][9] = f0;      G[tid][10] = f1;     G[tid][11] = f2;
        float dx = nx - cx, dy = ny - cy, dz = nz - cz;
        G[tid][12] = sqrtf(fmaxf(dx*dx + dy*dy + dz*dz, 1e-12f));
    }
    __syncthreads();
    float w[13];
    #pragma unroll
    for (int c = 0; c < 13; ++c) w[c] = W[tid*13 + c];
    float sc = bng[tid] * BN_INVS, bb = bnb[tid];
    float best = -FLT_MAX;
    for (int k = 0; k < KNN; ++k) {
        float acc = 0.0f;
        #pragma unroll
        for (int c = 0; c < 13; ++c) acc += w[c] * G[k][c];
        float y = acc * sc + bb;
        y = y > 0.0f ? y : SLOPE * y;
        best = fmaxf(best, y);
    }
    out[((size_t)b*64 + tid)*N + n] = best;
}

// ---------- farthest point sampling: one block per batch ----------
__global__ __launch_bounds__(256)
void fps_kernel(const float* __restrict__ pts, int* __restrict__ id, int N, int S) {
    const int b = blockIdx.x, tid = threadIdx.x;
    __shared__ float px[1024], py[1024], pz[1024], dist[1024];
    __shared__ float rv[256];
    __shared__ int   ri[256];
    __shared__ int   lastS;
    for (int m = tid; m < N; m += 256) {
        px[m] = pts[(b*3+0)*N + m];
        py[m] = pts[(b*3+1)*N + m];
        pz[m] = pts[(b*3+2)*N + m];
        dist[m] = 1e10f;
    }
    if (tid == 0) { id[b*S] = 0; lastS = 0; }
    __syncthreads();
    for (int s = 1; s < S; ++s) {
        int last = lastS;
        float lx = px[last], ly = py[last], lz = pz[last];
        float best = -FLT_MAX; int bi = 0x7fffffff;
        for (int m = tid; m < N; m += 256) {
            float dx = px[m]-lx, dy = py[m]-ly, dz = pz[m]-lz;
            float nd = fminf(dist[m], dx*dx + dy*dy + dz*dz);
            dist[m] = nd;
            if (nd > best) { best = nd; bi = m; }
        }
        rv[tid] = best; ri[tid] = bi;
        __syncthreads();
        for (int r = 128; r > 0; r >>= 1) {
            if (tid < r) {
                float vo = rv[tid+r]; int io = ri[tid+r];
                if (vo > rv[tid] || (vo == rv[tid] && io < ri[tid])) {
                    rv[tid] = vo; ri[tid] = io;
                }
            }
            __syncthreads();
        }
        if (tid == 0) { lastS = ri[0]; id[b*S + s] = ri[0]; }
        __syncthreads();
    }
}

// ---------- gather: out[b,c,m] = in[b,c,id[b,m]] ----------
__global__ void gather_kernel(const float* __restrict__ in, const int* __restrict__ id,
                              float* __restrict__ out, int C, int N, int M, int B) {
    int t = blockIdx.x * 256 + threadIdx.x;
    if (t >= B*C*M) return;
    int m = t % M, c = (t / M) % C, b = t / (M*C);
    out[((size_t)b*C + c)*M + m] = in[((size_t)b*C + c)*N + id[b*M + m]];
}

// ---------- channel concat: [a(64), b2(64), c3(128)] ----------
__global__ void concat3_kernel(const float* __restrict__ a, const float* __restrict__ b2,
                               const float* __restrict__ c3, float* __restrict__ out,
                               int N, int B) {
    int t = blockIdx.x * 256 + threadIdx.x;
    if (t >= B*256*N) return;
    int n = t % N, c = (t / N) % 256, b = t / (N*256);
    float v;
    if      (c < 64)  v = a [((size_t)b*64  + c      )*N + n];
    else if (c < 128) v = b2[((size_t)b*64  + (c-64) )*N + n];
    else              v = c3[((size_t)b*128 + (c-128))*N + n];
    out[((size_t)b*256 + c)*N + n] = v;
}

// ---------- fused group-gather + 256->Cout GEMM (f32 WMMA) + BN + lrelu + max-k ----------
#define CG_P    4
#define CG_COLS (CG_P*KNN)   // 80 columns = 5 full 16-col tiles
#define CG_LDB  68           // padded K-chunk stride (bank-conflict free)

template<int MTILES>   // Cout = MTILES*16 : 4 for conv2 (64), 8 for conv3 (128)
__global__ __launch_bounds__(256)
void conv_group_wmma(const float* __restrict__ X,       // (B,64,N)
                     const int* __restrict__ idx_eu,    // (B,N,20)
                     const int* __restrict__ idx_ei,
                     const float* __restrict__ W,       // (Cout,256)
                     const float* __restrict__ bng, const float* __restrict__ bnb,
                     float* __restrict__ out,           // (B,Cout,N)
                     int N) {
    constexpr int COUT   = MTILES * 16;
    constexpr int NTASKS = MTILES * (CG_COLS/16);     // 20 or 40
    constexpr int JMAX   = (NTASKS + 7) / 8;          // 3 or 5 (compile-time!)

    const int b = blockIdx.y;
    const int nbase = blockIdx.x * CG_P;
    const int tid = threadIdx.x;
    const int wave = tid >> 5, lane = tid & 31;

    __shared__ float Bls[CG_COLS * CG_LDB];
    __shared__ int   eu[CG_COLS], ei[CG_COLS];
    __shared__ int   YmaxI[CG_P * COUT];

    for (int t = tid; t < CG_P*COUT; t += 256) YmaxI[t] = f2ord(-3.0e38f);
    if (tid < CG_COLS) {
        int n = nbase + tid / KNN;
        eu[tid] = idx_eu[((size_t)b*N + n)*KNN + (tid % KNN)];
        ei[tid] = idx_ei[((size_t)b*N + n)*KNN + (tid % KNN)];
    }

    // accumulators: compile-time count & constant indices -> stay in VGPRs
    v8f acc[JMAX];
    #pragma unroll
    for (int j = 0; j < JMAX; ++j) acc[j] = v8f{0,0,0,0,0,0,0,0};

    const int lrow  = lane & 15;
    const int khalf = (lane < 16) ? 0 : 2;

    for (int kc = 0; kc < 4; ++kc) {       // 4 K-chunks of 64 = the 4 sections
        __syncthreads();
        // gather this 64-row section of all 80 columns into LDS
        for (int t = tid; t < CG_COLS*64; t += 256) {
            int c = t >> 6, r = t & 63;
            int n = nbase + c / KNN;
            int e = (kc < 2) ? eu[c] : ei[c];
            float xv = X[((size_t)b*64 + r)*N + e];
            if ((kc & 1) == 0) xv -= X[((size_t)b*64 + r)*N + n];  // diff sections
            Bls[c*CG_LDB + r] = xv;
        }
        __syncthreads();
        #pragma unroll
        for (int j = 0; j < JMAX; ++j) {
            const int t = wave + 8*j;          // wave-uniform guard
            if (t < NTASKS) {
                const int mt = t / 5, ct = t % 5;
                const float* wrow = W + (size_t)((mt<<4) + lrow)*256 + kc*64 + khalf;
                const float* bcol = Bls + (ct*16 + lrow)*CG_LDB + khalf;
                v8f a_ = acc[j];
                #pragma unroll
                for (int ks = 0; ks < 16; ++ks) {
                    v2f av = *(const v2f*)(wrow + 4*ks);   // A 16x4 frag (M=lrow, K pair)
                    v2f bv = *(const v2f*)(bcol + 4*ks);   // B 4x16 frag (N=lrow, K pair)
                    a_ = __builtin_amdgcn_wmma_f32_16x16x4_f32(
                            false, av, false, bv, (short)0, a_, false, false);
                }
                acc[j] = a_;
            }
        }
    }
    // epilogue: BN + lrelu + max over k via monotonic-int LDS atomic max
    #pragma unroll
    for (int j = 0; j < JMAX; ++j) {
        const int t = wave + 8*j;
        if (t < NTASKS) {
            const int mt = t / 5, ct = t % 5;
            const int col = ct*16 + lrow;
            const int p = col / KNN;
            #pragma unroll
            for (int r = 0; r < 8; ++r) {
                int o = (mt<<4) + ((lane < 16) ? r : r + 8);
                float y = acc[j][r] * (bng[o]*BN_INVS) + bnb[o];
                y = y > 0.0f ? y : SLOPE * y;
                atomicMax(&YmaxI[p*COUT + o], f2ord(y));
            }
        }
    }
    __syncthreads();
    for (int t = tid; t < CG_P*COUT; t += 256) {
        int p = t / COUT, o = t % COUT;
        out[((size_t)b*COUT + o)*N + nbase + p] = ord2f(YmaxI[p*COUT + o]);
    }
}

// ---------- conv5 GEMM (f32 WMMA) + BN + lrelu, fused global max/mean pooling ----------
__global__ void pool_init_kernel(int* __restrict__ pmax, float* __restrict__ psum, int total) {
    int t = blockIdx.x * 256 + threadIdx.x;
    if (t < total) { pmax[t] = f2ord(-3.0e38f); psum[t] = 0.0f; }
}

__global__ __launch_bounds__(256)
void conv5_pool_wmma(const float* __restrict__ W,     // (M,256)
                     const float* __restrict__ Hin,   // (B,256,N)
                     const float* __restrict__ bng, const float* __restrict__ bnb,
                     int* __restrict__ pmax, float* __restrict__ psum,
                     int N, int M) {
    const int b = blockIdx.y;
    const int lane = threadIdx.x & 31;
    const int ntTiles = N >> 4;
    const int tile = blockIdx.x * 8 + (threadIdx.x >> 5);
    if (tile >= (M >> 4) * ntTiles) return;   // wave-uniform
    const int mt = tile / ntTiles, nt = tile % ntTiles;
    const int lrow  = lane & 15;
    const int khalf = (lane < 16) ? 0 : 2;
    const int ncol  = (nt << 4) + lrow;
    const float* wrow = W + (size_t)((mt<<4) + lrow)*256 + khalf;
    const float* hb   = Hin + (size_t)b*256*N + ncol;
    v8f acc = {0,0,0,0,0,0,0,0};
    #pragma unroll 4
    for (int ks = 0; ks < 64; ++ks) {
        int k = 4*ks + khalf;
        v2f av = *(const v2f*)(wrow + 4*ks);
        v2f bv;
        bv.x = hb[(size_t)k*N];
        bv.y = hb[(size_t)(k+1)*N];
        acc = __builtin_amdgcn_wmma_f32_16x16x4_f32(
                false, av, false, bv, (short)0, acc, false, false);
    }
    #pragma unroll
    for (int r = 0; r < 8; ++r) {
        int o = (mt<<4) + ((lane < 16) ? r : r + 8);
        float y = acc[r] * (bng[o]*BN_INVS) + bnb[o];
        y = y > 0.0f ? y : SLOPE * y;
        atomicMax(&pmax[b*M + o], f2ord(y));
        atomicAdd(&psum[b*M + o], y);
    }
}

__global__ void pool_finalize_kernel(const int* __restrict__ pmax, const float* __restrict__ psum,
                                     float* __restrict__ pooled, int M, int N, int B) {
    int t = blockIdx.x * 256 + threadIdx.x;
    if (t >= B*M) return;
    int b = t / M, c = t % M;
    pooled[(size_t)b*2*M + c]     = ord2f(pmax[t]);
    pooled[(size_t)b*2*M + M + c] = psum[t] / (float)N;
}

// ---------- MLP head ----------
// mode 0: BN+lrelu (no bias); mode 1: +bias then BN+lrelu; mode 2: +bias only
__global__ void lin_kernel(const float* __restrict__ W, const float* __restrict__ bias,
                           const float* __restrict__ bng, const float* __restrict__ bnb,
                           const float* __restrict__ in, float* __restrict__ out,
                           int In, int Out, int mode) {
    int b = blockIdx.x, o = threadIdx.x;
    if (o >= Out) return;
    float acc = 0.0f;
    for (int i = 0; i < In; ++i) acc += W[(size_t)o*In + i] * in[(size_t)b*In + i];
    if (mode == 2) { out[b*Out + o] = acc + bias[o]; return; }
    if (mode == 1) acc += bias[o];
    float y = acc * BN_INVS * bng[o] + bnb[o];
    out[b*Out + o] = y > 0.0f ? y : SLOPE * y;
}

// ---------- host launch ----------
extern "C" void kernel_launch(void* const* d_in, const int* in_sizes, int n_in,
                              void* d_out, int out_size, void* d_ws, size_t ws_size,
                              hipStream_t stream) {
    const int B  = 16;
    const int N1 = in_sizes[0] / (B * 3);
    const int N2 = N1 / 2, N3 = N1 / 4;

    const float* x       = (const float*)d_in[0];
    const float* conv1_w = (const float*)d_in[1];
    const float* bn1g = (const float*)d_in[2],  *bn1b = (const float*)d_in[3];
    const float* conv2_w = (const float*)d_in[4];
    const float* bn2g = (const float*)d_in[5],  *bn2b = (const float*)d_in[6];
    const float* conv3_w = (const float*)d_in[7];
    const float* bn3g = (const float*)d_in[8],  *bn3b = (const float*)d_in[9];
    const float* conv5_w = (const float*)d_in[10];
    const float* bn5g = (const float*)d_in[11], *bn5b = (const float*)d_in[12];
    const float* lin1_w = (const float*)d_in[13];
    const float* bn6g = (const float*)d_in[14], *bn6b = (const float*)d_in[15];
    const float* lin2_w = (const float*)d_in[16], *lin2_b = (const float*)d_in[17];
    const float* bn7g = (const float*)d_in[18], *bn7b = (const float*)d_in[19];
    const float* lin3_w = (const float*)d_in[20], *lin3_b = (const float*)d_in[21];

    char* wsp = (char*)d_ws;
    auto carve = [&](size_t bytes) {
        void* p = (void*)wsp;
        wsp += (bytes + 255) & ~(size_t)255;
        return p;
    };
    float* eig1    = (float*)carve((size_t)B*3*N1*4);
    int*   idx_eu1 = (int*)  carve((size_t)B*N1*KNN*4);
    int*   idx_ei1 = (int*)  carve((size_t)B*N1*KNN*4);
    float* x1      = (float*)carve((size_t)B*64*N1*4);
    int*   id2     = (int*)  carve((size_t)B*N2*4);
    float* pts2    = (float*)carve((size_t)B*3*N2*4);
    float* x1d     = (float*)carve((size_t)B*64*N2*4);
    float* eig2    = (float*)carve((size_t)B*3*N2*4);
    int*   idx_eu2 = (int*)  carve((size_t)B*N2*KNN*4);
    int*   idx_ei2 = (int*)  carve((size_t)B*N2*KNN*4);
    float* x2      = (float*)carve((size_t)B*64*N2*4);
    int*   id3     = (int*)  carve((size_t)B*N3*4);
    float* pts3    = (float*)carve((size_t)B*3*N3*4);
    float* x2d     = (float*)carve((size_t)B*64*N3*4);
    float* x1d3    = (float*)carve((size_t)B*64*N3*4);
    float* eig3    = (float*)carve((size_t)B*3*N3*4);
    int*   idx_eu3 = (int*)  carve((size_t)B*N3*KNN*4);
    int*   idx_ei3 = (int*)  carve((size_t)B*N3*KNN*4);
    float* x3      = (float*)carve((size_t)B*128*N3*4);
    float* hcat    = (float*)carve((size_t)B*256*N3*4);
    int*   pmax    = (int*)  carve((size_t)B*1024*4);
    float* psum    = (float*)carve((size_t)B*1024*4);
    float* pooled  = (float*)carve((size_t)B*2048*4);
    float* y1      = (float*)carve((size_t)B*512*4);
    float* y2      = (float*)carve((size_t)B*256*4);

    // ---- level 1 ----
    knn3_kernel<<<dim3(N1,B), 256, 0, stream>>>(x, idx_eu1, N1);
    eig_kernel<<<(B*N1+255)/256, 256, 0, stream>>>(x, idx_eu1, eig1, N1, B);
    knn3_kernel<<<dim3(N1,B), 256, 0, stream>>>(eig1, idx_ei1, N1);
    conv1_kernel<<<dim3(N1,B), 64, 0, stream>>>(x, eig1, idx_eu1, idx_ei1,
                                                conv1_w, bn1g, bn1b, x1, N1);
    // ---- level 2 ----
    fps_kernel<<<B, 256, 0, stream>>>(x, id2, N1, N2);
    gather_kernel<<<(B*3*N2+255)/256, 256, 0, stream>>>(x,  id2, pts2, 3,  N1, N2, B);
    gather_kernel<<<(B*64*N2+255)/256, 256, 0, stream>>>(x1, id2, x1d, 64, N1, N2, B);
    knn3_kernel<<<dim3(N2,B), 256, 0, stream>>>(pts2, idx_eu2, N2);
    eig_kernel<<<(B*N2+255)/256, 256, 0, stream>>>(pts2, idx_eu2, eig2, N2, B);
    knn3_kernel<<<dim3(N2,B), 256, 0, stream>>>(eig2, idx_ei2, N2);
    conv_group_wmma<4><<<dim3(N2/CG_P,B), 256, 0, stream>>>(x1d, idx_eu2, idx_ei2,
                                                            conv2_w, bn2g, bn2b, x2, N2);
    // ---- level 3 ----
    fps_kernel<<<B, 256, 0, stream>>>(pts2, id3, N2, N3);
    gather_kernel<<<(B*3*N3+255)/256, 256, 0, stream>>>(pts2, id3, pts3, 3,  N2, N3, B);
    gather_kernel<<<(B*64*N3+255)/256, 256, 0, stream>>>(x2,  id3, x2d,  64, N2, N3, B);
    gather_kernel<<<(B*64*N3+255)/256, 256, 0, stream>>>(x1d, id3, x1d3, 64, N2, N3, B);
    knn3_kernel<<<dim3(N3,B), 256, 0, stream>>>(pts3, idx_eu3, N3);
    eig_kernel<<<(B*N3+255)/256, 256, 0, stream>>>(pts3, idx_eu3, eig3, N3, B);
    knn3_kernel<<<dim3(N3,B), 256, 0, stream>>>(eig3, idx_ei3, N3);
    conv_group_wmma<8><<<dim3(N3/CG_P,B), 256, 0, stream>>>(x2d, idx_eu3, idx_ei3,
                                                            conv3_w, bn3g, bn3b, x3, N3);
    // ---- embedding + pooled head ----
    concat3_kernel<<<(B*256*N3+255)/256, 256, 0, stream>>>(x1d3, x2d, x3, hcat, N3, B);
    pool_init_kernel<<<(B*1024+255)/256, 256, 0, stream>>>(pmax, psum, B*1024);
    conv5_pool_wmma<<<dim3(((1024/16)*(N3/16))/8, B), 256, 0, stream>>>(
        conv5_w, hcat, bn5g, bn5b, pmax, psum, N3, 1024);
    pool_finalize_kernel<<<(B*1024+255)/256, 256, 0, stream>>>(pmax, psum, pooled, 1024, N3, B);

    lin_kernel<<<B, 512, 0, stream>>>(lin1_w, nullptr, bn6g, bn6b, pooled, y1, 2048, 512, 0);
    lin_kernel<<<B, 256, 0, stream>>>(lin2_w, lin2_b, bn7g, bn7b, y1, y2, 512, 256, 1);
    lin_kernel<<<B, 64, 0, stream>>>(lin3_w, lin3_b, nullptr, nullptr, y2,
                                     (float*)d_out, 256, 40, 2);
    (void)n_in; (void)out_size; (void)ws_size;
}